// ConfidenceLoss_51041391345678
// MI455X (gfx1250) — compile-verified
//
#include <hip/hip_runtime.h>
#include <stdint.h>

// ---------------- problem constants ----------------
#define TOTAL      558848          // B*D = 64*8732
#define CLS        21
#define TILE       256             // boxes per workgroup
#define NBLK       (TOTAL / TILE)  // 2183, exact (no tail)
#define NBLK2      1024            // blocks for the selection pass
#define TILE_BYTES (TILE * CLS * 4) // 21504 bytes per tile per tensor

// ---------------- workspace layout (uint32 units) ----------------
#define SC_N        0   // positive count
#define SC_POSLOSS  1   // float bits
#define SC_K        2
#define SC_T        3   // level-1 threshold bin (0x10000 = sentinel "none")
#define SC_REM      4   // remaining count to take from bin T
#define HIST1_OFF   8
#define HIST2_OFF   (HIST1_OFF + 65536)
#define BPOS_OFF    (HIST2_OFF + 65536)     // NBLK float partials (bits)
#define BCNT_OFF    (BPOS_OFF + NBLK)       // NBLK uint partials
#define BSUM2_OFF   (BCNT_OFF + NBLK)       // NBLK2 float partials (bits)
#define KEYS_OFF    (BSUM2_OFF + NBLK2)     // TOTAL keys

// ---------------- CDNA5 async copy to LDS ----------------
#if defined(__has_builtin)
#if __has_builtin(__builtin_amdgcn_global_load_async_to_lds_b128)
#define HAVE_ASYNC_BUILTIN 1
#endif
#if __has_builtin(__builtin_amdgcn_s_wait_asynccnt)
#define HAVE_WAIT_BUILTIN 1
#endif
#endif

typedef int v4i __attribute__((ext_vector_type(4)));
typedef __attribute__((address_space(1))) v4i  as1_v4i;   // global int4
typedef __attribute__((address_space(3))) v4i  as3_v4i;   // LDS int4
typedef __attribute__((address_space(3))) char as3_char;

__device__ __forceinline__ void async_wait0() {
#ifdef HAVE_WAIT_BUILTIN
  __builtin_amdgcn_s_wait_asynccnt(0);
#else
  asm volatile("s_wait_asynccnt 0" ::: "memory");
#endif
}

// copy TILE_BYTES bytes (multiple of 16, both 16B aligned) global -> LDS
__device__ __forceinline__ void copy_tile_async(const float* __restrict__ g,
                                                float* l, int tid) {
#ifdef HAVE_ASYNC_BUILTIN
  char* gb = (char*)g;
  as3_char* lb = (as3_char*)l;   // addrspacecast generic -> LDS
  #pragma unroll
  for (int off = tid * 16; off < TILE_BYTES; off += 256 * 16) {
    __builtin_amdgcn_global_load_async_to_lds_b128(
        (as1_v4i*)(gb + off), (as3_v4i*)(lb + off), 0, 0);
  }
#else
  // inline-asm fallback: vdst = LDS byte address, vaddr = 64-bit global address
  as3_char* lb = (as3_char*)l;
  unsigned lbase = (unsigned)(size_t)lb;
  unsigned long long gbase = (unsigned long long)(size_t)g;
  for (int off = tid * 16; off < TILE_BYTES; off += 256 * 16) {
    unsigned laddr = lbase + (unsigned)off;
    unsigned long long gaddr = gbase + (unsigned long long)off;
    asm volatile("global_load_async_to_lds_b128 %0, %1, off"
                 :: "v"(laddr), "v"(gaddr) : "memory");
  }
#endif
}

// ---------------- kernel 0: zero the two histograms ----------------
__global__ void k_init(unsigned* __restrict__ ws) {
  int stride = gridDim.x * blockDim.x;
  for (int i = blockIdx.x * blockDim.x + threadIdx.x; i < 2 * 65536; i += stride)
    ws[HIST1_OFF + i] = 0u;
}

// ---------------- kernel 1: stream, per-box math, keys + hist1 ----------------
__global__ __launch_bounds__(256) void k_main(const float* __restrict__ pred,
                                              const float* __restrict__ gts,
                                              const unsigned char* __restrict__ pos,
                                              unsigned* __restrict__ ws) {
  __shared__ __align__(16) float lp[TILE * CLS];
  __shared__ __align__(16) float lg[TILE * CLS];
  __shared__ float    redf[256];
  __shared__ unsigned redc[256];

  const int tid = threadIdx.x;
  const long long base = (long long)blockIdx.x * (TILE * CLS);

  copy_tile_async(pred + base, lp, tid);
  copy_tile_async(gts + base, lg, tid);
  async_wait0();          // this wave's async copies landed in LDS
  __syncthreads();        // all waves' copies landed

  const float* x = &lp[tid * CLS];
  const float* g = &lg[tid * CLS];

  float m = x[0];
  #pragma unroll
  for (int c = 1; c < CLS; ++c) m = fmaxf(m, x[c]);
  float s1 = 0.f, dg = 0.f, sg = 0.f;
  #pragma unroll
  for (int c = 0; c < CLS; ++c) {
    s1 += __expf(x[c] - m);
    dg += g[c] * x[c];
    sg += g[c];
  }
  const float lse = m + __logf(s1);

  const float posC = sg * lse - dg;                       // sum_c g_c*(lse - x_c)
  const float bg   = fmaxf(g[CLS - 1] * (lse - x[CLS - 1]), 0.0f);

  const int box = blockIdx.x * TILE + tid;
  const bool isPos = pos[box] != 0;

  // candidate key: positives -> 0 (value +0.0, can only pad the zero region)
  const unsigned key = isPos ? 0u : __float_as_uint(bg);
  ws[KEYS_OFF + box] = key;
  atomicAdd(&ws[HIST1_OFF + (key >> 16)], 1u);

  // deterministic block reduction of positive loss / count
  redf[tid] = isPos ? posC : 0.0f;
  redc[tid] = isPos ? 1u : 0u;
  __syncthreads();
  for (int s = 128; s > 0; s >>= 1) {
    if (tid < s) { redf[tid] += redf[tid + s]; redc[tid] += redc[tid + s]; }
    __syncthreads();
  }
  if (tid == 0) {
    ws[BPOS_OFF + blockIdx.x] = __float_as_uint(redf[0]);
    ws[BCNT_OFF + blockIdx.x] = redc[0];
  }
}

// ---------------- kernel 2: reduce partials, pick level-1 threshold bin ----------------
__global__ __launch_bounds__(1024) void k_scan1(unsigned* __restrict__ ws) {
  __shared__ unsigned csum[1024];
  __shared__ unsigned ccnt[1024];
  __shared__ float    cpos[1024];
  const int tid = threadIdx.x;

  unsigned cnt = 0; float pl = 0.f;
  for (int i = tid; i < NBLK; i += 1024) {
    cnt += ws[BCNT_OFF + i];
    pl  += __uint_as_float(ws[BPOS_OFF + i]);
  }
  ccnt[tid] = cnt; cpos[tid] = pl;

  unsigned h = 0;
  #pragma unroll 4
  for (int j = 0; j < 64; ++j) h += ws[HIST1_OFF + tid * 64 + j];
  csum[tid] = h;
  __syncthreads();
  for (int s = 512; s > 0; s >>= 1) {
    if (tid < s) { ccnt[tid] += ccnt[tid + s]; cpos[tid] += cpos[tid + s]; }
    __syncthreads();
  }

  if (tid == 0) {
    const unsigned N = ccnt[0];
    long long k = 3LL * (long long)N;
    const long long other = (long long)TOTAL - (long long)N;
    if (other < k) k = other;

    unsigned t = 0x10000u, rem = 0u;
    if (k > 0) {
      long long cum = 0;
      for (int c = 1023; c >= 0; --c) {
        if (cum + (long long)csum[c] >= k) {
          for (int b = c * 64 + 63;; --b) {
            const unsigned hb = ws[HIST1_OFF + b];
            if (cum + (long long)hb >= k) { t = (unsigned)b; rem = (unsigned)(k - cum); break; }
            cum += hb;
          }
          break;
        }
        cum += csum[c];
      }
    }
    ws[SC_N] = N;
    ws[SC_POSLOSS] = __float_as_uint(cpos[0]);
    ws[SC_K] = (unsigned)k;
    ws[SC_T] = t;
    ws[SC_REM] = rem;
  }
}

// ---------------- kernel 3: sum above threshold bin + low-16 histogram of tie bin ----------------
__global__ __launch_bounds__(256) void k_pass2(unsigned* __restrict__ ws) {
  __shared__ float redf[256];
  const unsigned t = ws[SC_T];
  float s = 0.f;
  const int stride = gridDim.x * 256;
  for (int i = blockIdx.x * 256 + threadIdx.x; i < TOTAL; i += stride) {
    const unsigned key = ws[KEYS_OFF + i];
    const unsigned hi = key >> 16;
    if (hi > t)       s += __uint_as_float(key);
    else if (hi == t) atomicAdd(&ws[HIST2_OFF + (key & 0xFFFFu)], 1u);
  }
  redf[threadIdx.x] = s;
  __syncthreads();
  for (int r = 128; r > 0; r >>= 1) {
    if (threadIdx.x < r) redf[threadIdx.x] += redf[threadIdx.x + r];
    __syncthreads();
  }
  if (threadIdx.x == 0) ws[BSUM2_OFF + blockIdx.x] = __float_as_uint(redf[0]);
}

// ---------------- kernel 4: resolve ties exactly, produce the scalar ----------------
__global__ __launch_bounds__(1024) void k_scan2(unsigned* __restrict__ ws,
                                                float* __restrict__ out) {
  __shared__ unsigned csum[1024];
  __shared__ float    fred[1024];
  __shared__ unsigned sJ, sRem2;
  __shared__ float    sAbove;
  const int tid = threadIdx.x;

  unsigned h = 0;
  #pragma unroll 4
  for (int j = 0; j < 64; ++j) h += ws[HIST2_OFF + tid * 64 + j];
  csum[tid] = h;
  fred[tid] = __uint_as_float(ws[BSUM2_OFF + tid]);   // exactly NBLK2 == 1024 partials
  __syncthreads();
  for (int s = 512; s > 0; s >>= 1) {
    if (tid < s) fred[tid] += fred[tid + s];
    __syncthreads();
  }
  if (tid == 0) {
    sAbove = fred[0];
    const unsigned rem = ws[SC_REM];
    unsigned jst = 0x10000u, rem2 = 0u;
    if (rem > 0) {
      long long cum = 0;
      for (int c = 1023; c >= 0; --c) {
        if (cum + (long long)csum[c] >= (long long)rem) {
          for (int b = c * 64 + 63;; --b) {
            const unsigned hb = ws[HIST2_OFF + b];
            if (cum + (long long)hb >= (long long)rem) {
              jst = (unsigned)b; rem2 = (unsigned)(rem - cum); break;
            }
            cum += hb;
          }
          break;
        }
        cum += csum[c];
      }
    }
    sJ = jst; sRem2 = rem2;
  }
  __syncthreads();

  const unsigned jst = sJ;
  const unsigned t = ws[SC_T];
  float w = 0.f;
  for (int j = 0; j < 64; ++j) {
    const unsigned bin = (unsigned)(tid * 64 + j);
    if (bin > jst) {
      const unsigned c = ws[HIST2_OFF + bin];
      if (c) w += (float)c * __uint_as_float((t << 16) | bin);
    }
  }
  __syncthreads();
  fred[tid] = w;
  __syncthreads();
  for (int s = 512; s > 0; s >>= 1) {
    if (tid < s) fred[tid] += fred[tid + s];
    __syncthreads();
  }
  if (tid == 0) {
    float negTie = 0.f;
    if (jst <= 0xFFFFu && sRem2)
      negTie = (float)sRem2 * __uint_as_float((t << 16) | jst);
    const float posLoss = __uint_as_float(ws[SC_POSLOSS]);
    const float Nf = (float)ws[SC_N];
    out[0] = (posLoss + sAbove + fred[0] + negTie) / Nf;
  }
}

// ---------------- host launcher ----------------
extern "C" void kernel_launch(void* const* d_in, const int* in_sizes, int n_in,
                              void* d_out, int out_size, void* d_ws, size_t ws_size,
                              hipStream_t stream) {
  (void)in_sizes; (void)n_in; (void)out_size; (void)ws_size;
  const float* pred = (const float*)d_in[0];
  const float* gts  = (const float*)d_in[1];
  const unsigned char* pos = (const unsigned char*)d_in[2];  // jax bool -> 1 byte
  unsigned* ws = (unsigned*)d_ws;
  float* out = (float*)d_out;

  k_init <<<128,  256, 0, stream>>>(ws);
  k_main <<<NBLK, 256, 0, stream>>>(pred, gts, pos, ws);
  k_scan1<<<1,   1024, 0, stream>>>(ws);
  k_pass2<<<NBLK2, 256, 0, stream>>>(ws);
  k_scan2<<<1,   1024, 0, stream>>>(ws, out);
}